// BilinearDecoder_47571057770814
// MI455X (gfx1250) — compile-verified
//
#include <hip/hip_runtime.h>
#include <hip/hip_bf16.h>

typedef __attribute__((ext_vector_type(2))) float v2f;
typedef __attribute__((ext_vector_type(4))) float v4f;
typedef __attribute__((ext_vector_type(8))) float v8f;

#define DIM 64

// ---------------------------------------------------------------------------
// Kernel 1: zW = z @ W  via V_WMMA_F32_16X16X4_F32 (fp32 in, fp32 acc)
// One wave32 computes one 16x16 tile of zW. K loop = 64/4 = 16 WMMA steps.
//
// A (16x4 f32) lane layout:  lanes 0-15 -> M=lane, VGPR0=K0,VGPR1=K1
//                            lanes 16-31 -> M=lane-16, VGPR0=K2,VGPR1=K3
// B (4x16 f32) lane layout (mirror): lanes 0-15 -> N=lane, K={0,1};
//                                    lanes 16-31 -> N=lane-16, K={2,3}
// D (16x16 f32): VGPR r -> row M = r + 8*(lane>=16), col N = lane&15
// ---------------------------------------------------------------------------
__global__ void __launch_bounds__(256)
zw_gemm_wmma_kernel(const float* __restrict__ z,
                    const float* __restrict__ W,
                    float* __restrict__ zW,
                    int nTiles)            // nTiles = (N/16)*4
{
    const int wave   = threadIdx.x >> 5;
    const int tile   = blockIdx.x * 8 + wave;
    if (tile >= nTiles) return;            // whole-wave guard (EXEC stays all-1)

    const int colT   = tile & 3;           // 4 column tiles of 16 (DIM=64)
    const int rowT   = tile >> 2;
    const int row0   = rowT * 16;
    const int col0   = colT * 16;

    const int lane   = threadIdx.x & 31;
    const int half   = lane >> 4;          // 0: K={k,k+1}; 1: K={k+2,k+3}
    const int l      = lane & 15;          // M index (A) / N index (B)

    v8f acc = {};

    #pragma unroll
    for (int k = 0; k < DIM; k += 4) {
        const int kk = k + half * 2;
        // A: z[row0+l][kk..kk+1]  (8B aligned: kk is even)
        v2f a = *(const v2f*)(z + (size_t)(row0 + l) * DIM + kk);
        // B: W[kk..kk+1][col0+l]  (column gather from 16KB W, L2-resident)
        v2f b;
        b.x = W[(size_t)kk       * DIM + col0 + l];
        b.y = W[(size_t)(kk + 1) * DIM + col0 + l];
        acc = __builtin_amdgcn_wmma_f32_16x16x4_f32(
                  /*neg_a=*/false, a, /*neg_b=*/false, b,
                  /*c_mod=*/(short)0, acc, /*reuse_a=*/false, /*reuse_b=*/false);
    }

    #pragma unroll
    for (int r = 0; r < 8; ++r) {
        zW[(size_t)(row0 + r + 8 * half) * DIM + col0 + l] = acc[r];
    }
}

// ---------------------------------------------------------------------------
// Kernel 2: per-edge gather-dot. One thread per edge; 16+16 b128 gathers.
// Working set (z + zW = 51 MB) is resident in the 192 MB L2, so these
// gathers hit L2; many in-flight b128 loads per wave hide the latency.
// ---------------------------------------------------------------------------
__global__ void __launch_bounds__(256)
edge_score_kernel(const float* __restrict__ zW,
                  const float* __restrict__ z,
                  const long long* __restrict__ edge_index,
                  const float* __restrict__ bias,
                  float* __restrict__ out,
                  int E)
{
    const int e = blockIdx.x * blockDim.x + threadIdx.x;
    if (e >= E) return;

    const long long s = edge_index[e];        // src row
    const long long d = edge_index[(size_t)E + e];  // dst row

    const v4f* __restrict__ a = (const v4f*)(zW + (size_t)s * DIM);
    const v4f* __restrict__ b = (const v4f*)(z  + (size_t)d * DIM);

    float acc0 = 0.f, acc1 = 0.f, acc2 = 0.f, acc3 = 0.f;
    #pragma unroll
    for (int i = 0; i < DIM / 4; ++i) {
        v4f av = a[i];
        v4f bv = b[i];
        acc0 = fmaf(av.x, bv.x, acc0);
        acc1 = fmaf(av.y, bv.y, acc1);
        acc2 = fmaf(av.z, bv.z, acc2);
        acc3 = fmaf(av.w, bv.w, acc3);
    }
    out[e] = (acc0 + acc1) + (acc2 + acc3) + bias[0];
}

// ---------------------------------------------------------------------------
extern "C" void kernel_launch(void* const* d_in, const int* in_sizes, int n_in,
                              void* d_out, int out_size, void* d_ws, size_t ws_size,
                              hipStream_t stream)
{
    const float*     z    = (const float*)d_in[0];       // [N, 64]
    const long long* ei   = (const long long*)d_in[1];   // [2, E] int64
    const float*     W    = (const float*)d_in[2];       // [64, 64]
    const float*     bias = (const float*)d_in[3];       // [1]
    float*           out  = (float*)d_out;               // [E]
    float*           zW   = (float*)d_ws;                // [N, 64] scratch

    const int nNodes = in_sizes[0] / DIM;
    const int E      = in_sizes[1] / 2;

    // Phase 1: zW = z @ W (WMMA). (N/16)*4 tiles, 8 waves per 256-thread block.
    const int nTiles = (nNodes / 16) * 4;
    const int gemmBlocks = (nTiles + 7) / 8;
    zw_gemm_wmma_kernel<<<gemmBlocks, 256, 0, stream>>>(z, W, zW, nTiles);

    // Phase 2: gather-dot over edges.
    const int edgeBlocks = (E + 255) / 256;
    edge_score_kernel<<<edgeBlocks, 256, 0, stream>>>(zW, z, ei, bias, out, E);
}